// NearestNeighbor_15530601742645
// MI455X (gfx1250) — compile-verified
//
#include <hip/hip_runtime.h>

typedef float v2f __attribute__((ext_vector_type(2)));
typedef float v4f __attribute__((ext_vector_type(4)));
typedef float v8f __attribute__((ext_vector_type(8)));

#define DIN 1056
#define TILE_M 16

__global__ void nn_init_ws(unsigned long long* ws) {
    if (threadIdx.x == 0) ws[0] = 0xFFFFFFFFFFFFFFFFull;
}

// One wave processes a tile of 16 db rows, 8 K-elements per step via two
// V_WMMA_F32_16X16X4_F32. K-mapping is permuted (dot products are K-order
// invariant) so each lane issues one b128 load per step:
//   lane L    (<16): row L, data k0..k0+3   -> WMMA1 hwK{0,1} = .xy, WMMA2 hwK{0,1} = .zw
//   lane L+16      : row L, data k0+4..k0+7 -> WMMA1 hwK{2,3} = .xy, WMMA2 hwK{2,3} = .zw
// B is the query packed with the identical per-lane K mapping, broadcast to all
// 16 columns => D[m][n] = dot(db_row_m, q) in every column.
// The K loop is software-pipelined (depth 2) so each wave keeps >=2 load pairs
// in flight while WMMAs of the oldest chunk execute.
__global__ __launch_bounds__(128) void nn_dist_kernel(
    const float* __restrict__ q, const float* __restrict__ db,
    int nRows, unsigned long long* __restrict__ ws) {
    const int lane = threadIdx.x & 31;
    const int wavesPerBlock = blockDim.x >> 5;
    const int waveId = blockIdx.x * wavesPerBlock + (threadIdx.x >> 5);
    const int totalWaves = gridDim.x * wavesPerBlock;
    const int nTiles = nRows / TILE_M;

    const int m    = lane & 15;        // row within tile
    const int koff = (lane >> 4) * 4;  // 0 for lanes 0-15, 4 for lanes 16-31

    float bestD = __builtin_inff();
    int   bestI = 0;

    for (int tile = waveId; tile < nTiles; tile += totalWaves) {
        const int rowBase = tile * TILE_M;
        const float* rowPtr = db + (size_t)(rowBase + m) * DIN + koff;
        const float* qPtr   = q + koff;

        v8f c = {};
        float ssq = 0.0f;

#define NN_STEP(aa, bb)                                                        \
    do {                                                                       \
        ssq = fmaf((aa).x, (aa).x, ssq);                                       \
        ssq = fmaf((aa).y, (aa).y, ssq);                                       \
        ssq = fmaf((aa).z, (aa).z, ssq);                                       \
        ssq = fmaf((aa).w, (aa).w, ssq);                                       \
        v2f aLo = {(aa).x, (aa).y}, bLo = {(bb).x, (bb).y};                    \
        v2f aHi = {(aa).z, (aa).w}, bHi = {(bb).z, (bb).w};                    \
        c = __builtin_amdgcn_wmma_f32_16x16x4_f32(                             \
                false, aLo, false, bLo, (short)0, c, false, false);            \
        c = __builtin_amdgcn_wmma_f32_16x16x4_f32(                             \
                false, aHi, false, bHi, (short)0, c, false, false);            \
    } while (0)

        // Software pipeline, depth 2 (DIN/8 = 132 steps >= 2).
        v4f a0 = *(const v4f*)(rowPtr + 0);
        v4f b0 = *(const v4f*)(qPtr + 0);
        v4f a1 = *(const v4f*)(rowPtr + 8);
        v4f b1 = *(const v4f*)(qPtr + 8);
        for (int k0 = 0; k0 + 16 < DIN; k0 += 8) {
            v4f a2 = *(const v4f*)(rowPtr + k0 + 16);
            v4f b2 = *(const v4f*)(qPtr + k0 + 16);
            NN_STEP(a0, b0);
            a0 = a1; b0 = b1;
            a1 = a2; b1 = b2;
        }
        NN_STEP(a0, b0);
        NN_STEP(a1, b1);
#undef NN_STEP

        // Row r total ssq = lane r + lane r+16; fold halves once.
        float ssqTot = ssq + __shfl_xor(ssq, 16, 32);

        // Extract 16 distances (wave-uniform) and fold into running min.
        // C/D layout: VGPR v -> row v in lanes 0-15, row v+8 in lanes 16-31.
#pragma unroll
        for (int v = 0; v < 8; ++v) {
            float dot_lo = __shfl(c[v], 0, 32);
            float dot_hi = __shfl(c[v], 16, 32);
            float ssq_lo = __shfl(ssqTot, v, 32);
            float ssq_hi = __shfl(ssqTot, v + 8, 32);
            float dLo = fmaf(-2.0f, dot_lo, ssq_lo);
            float dHi = fmaf(-2.0f, dot_hi, ssq_hi);
            int iLo = rowBase + v;
            int iHi = rowBase + v + 8;
            if (dLo < bestD) { bestD = dLo; bestI = iLo; }
            if (dHi < bestD) { bestD = dHi; bestI = iHi; }
        }
    }

    // Remainder rows (nRows % 16), handled by wave 0 with plain VALU + reduce.
    if (waveId == 0) {
        for (int r = (nRows / TILE_M) * TILE_M; r < nRows; ++r) {
            const float* row = db + (size_t)r * DIN;
            float s = 0.0f;
            for (int k = lane; k < DIN; k += 32) {
                float d = row[k] - q[k];
                s = fmaf(d, d, s);
            }
            for (int off = 16; off > 0; off >>= 1)
                s += __shfl_xor(s, off, 32);
            if (s < bestD) { bestD = s; bestI = r; }
        }
    }

    if (lane == 0) {
        // Monotone float->uint key packed with index: u64 atomicMin performs a
        // global argmin with first-occurrence tie-breaking.
        unsigned ub  = __float_as_uint(bestD);
        unsigned key = (ub & 0x80000000u) ? ~ub : (ub | 0x80000000u);
        unsigned long long packed =
            ((unsigned long long)key << 32) | (unsigned)bestI;
        atomicMin(ws, packed);
    }
}

__global__ void nn_gather(const float* __restrict__ pred, int dPred,
                          const unsigned long long* __restrict__ ws,
                          float* __restrict__ out) {
    unsigned idx = (unsigned)(ws[0] & 0xFFFFFFFFull);
    const float* src = pred + (size_t)idx * dPred;
    for (int i = blockIdx.x * blockDim.x + threadIdx.x; i < dPred;
         i += blockDim.x * gridDim.x)
        out[i] = src[i];
}

extern "C" void kernel_launch(void* const* d_in, const int* in_sizes, int n_in,
                              void* d_out, int out_size, void* d_ws, size_t ws_size,
                              hipStream_t stream) {
    const float* q    = (const float*)d_in[0];   // [1, 1056]
    const float* db   = (const float*)d_in[1];   // [N, 1056]
    const float* pred = (const float*)d_in[2];   // [N, 924]
    const int dIn   = in_sizes[0];               // 1056
    const int nRows = in_sizes[1] / dIn;         // 100000
    unsigned long long* ws = (unsigned long long*)d_ws;

    nn_init_ws<<<1, 32, 0, stream>>>(ws);

    const int nTiles = nRows / TILE_M;           // 6250
    const int wavesPerBlock = 4;                 // 128 threads
    int blocks = (nTiles + wavesPerBlock - 1) / wavesPerBlock;
    if (blocks < 1) blocks = 1;
    nn_dist_kernel<<<blocks, 128, 0, stream>>>(q, db, nRows, ws);

    nn_gather<<<4, 256, 0, stream>>>(pred, out_size, ws, (float*)d_out);
}